// TaskMoE_13288628813932
// MI455X (gfx1250) — compile-verified
//
#include <hip/hip_runtime.h>
#include <hip/hip_bf16.h>

// MoE top-2 forward for MI455X (gfx1250, wave32).
// T=8192 tokens, H=1024, E=8 experts, TOP_K=2.
// route -> bias-init -> two grouped-GEMM passes (top1, top2):
//   V_WMMA_F32_16X16X4_F32, double-buffered B tile staged in LDS via
//   GLOBAL_LOAD_ASYNC_TO_LDS (ASYNCcnt pipeline), gathered A rows,
//   gate-scaled RMW epilogue.

constexpr int Tt = 4 * 2048;   // tokens
constexpr int Hd = 1024;       // hidden
constexpr int NE = 8;          // experts

typedef __attribute__((ext_vector_type(2))) float v2f;
typedef __attribute__((ext_vector_type(8))) float v8f;

#if __has_builtin(__builtin_amdgcn_global_load_async_to_lds_b128) && \
    __has_builtin(__builtin_amdgcn_s_wait_asynccnt)
#define USE_ASYNC_LDS 1
#endif

// exact parameter pointee type from hipcc diagnostic: 'int __vector(4)'
typedef int v4i_ls __attribute__((__vector_size__(4 * sizeof(int))));
#define GASV(p) ((__attribute__((address_space(1))) v4i_ls*)(p))
#define LASV(p) ((__attribute__((address_space(3))) v4i_ls*)(p))

// ---------------------------------------------------------------- routing ---
// One wave (32 lanes) per token: 8 logits, normalized top-2 gates.
__global__ __launch_bounds__(256) void moe_route_k(
    const float* __restrict__ x, const float* __restrict__ Wg,
    int* __restrict__ counts, int* __restrict__ lists,
    int* __restrict__ topidx, float* __restrict__ gws)
{
    const int lane = threadIdx.x & 31;
    const int t = blockIdx.x * 8 + (threadIdx.x >> 5);

    float acc[NE];
#pragma unroll
    for (int e = 0; e < NE; ++e) acc[e] = 0.f;

    const float* xr = x + (size_t)t * Hd;
    for (int h = lane; h < Hd; h += 32) {
        const float xv = xr[h];
        const float4* wg = (const float4*)(Wg + (size_t)h * NE);
        const float4 w0 = wg[0], w1 = wg[1];
        acc[0] += xv * w0.x; acc[1] += xv * w0.y;
        acc[2] += xv * w0.z; acc[3] += xv * w0.w;
        acc[4] += xv * w1.x; acc[5] += xv * w1.y;
        acc[6] += xv * w1.z; acc[7] += xv * w1.w;
    }
#pragma unroll
    for (int e = 0; e < NE; ++e) {
#pragma unroll
        for (int off = 16; off > 0; off >>= 1)
            acc[e] += __shfl_xor(acc[e], off, 32);
    }

    if (lane == 0) {
        float l0 = -3.4e38f; int i0 = 0;       // first index wins ties
#pragma unroll
        for (int e = 0; e < NE; ++e)
            if (acc[e] > l0) { l0 = acc[e]; i0 = e; }
        float l1 = -3.4e38f; int i1 = 0;
#pragma unroll
        for (int e = 0; e < NE; ++e)
            if (e != i0 && acc[e] > l1) { l1 = acc[e]; i1 = e; }

        const float d  = __expf(l1 - l0);       // <= 1
        const float g0 = 1.f / (1.f + d);
        const float g1 = d / (1.f + d);

        const int p0 = atomicAdd(&counts[0 * NE + i0], 1);
        lists[(size_t)(0 * NE + i0) * Tt + p0] = t;
        const int p1 = atomicAdd(&counts[1 * NE + i1], 1);
        lists[(size_t)(1 * NE + i1) * Tt + p1] = t;

        topidx[t] = i0;  topidx[Tt + t] = i1;
        gws[t]    = g0;  gws[Tt + t]    = g1;
    }
}

// ------------------------------------------------------------ bias / init ---
// out[t,:] = g0*be[e0,:] + g1*be[e1,:]  (also overwrites poisoned d_out)
__global__ __launch_bounds__(256) void moe_bias_k(
    const float* __restrict__ be, const int* __restrict__ topidx,
    const float* __restrict__ gws, float* __restrict__ out)
{
    const int idx = blockIdx.x * 256 + threadIdx.x;  // float4 index
    const int t  = idx >> 8;                         // Hd/4 = 256 vec4 per row
    const int h4 = (idx & 255) << 2;
    const int i0 = topidx[t], i1 = topidx[Tt + t];
    const float g0 = gws[t], g1 = gws[Tt + t];
    const float4 b0 = *(const float4*)(be + (size_t)i0 * Hd + h4);
    const float4 b1 = *(const float4*)(be + (size_t)i1 * Hd + h4);
    float4 o;
    o.x = g0 * b0.x + g1 * b1.x;
    o.y = g0 * b0.y + g1 * b1.y;
    o.z = g0 * b0.z + g1 * b1.z;
    o.w = g0 * b0.w + g1 * b1.w;
    *(float4*)(out + (size_t)t * Hd + h4) = o;
}

// --------------------------------------------------- grouped WMMA GEMM ------
// Pass kp in {0,1}. Per expert e: out[tok,:] += gate * (x[tok,:] @ We[e]).
// WG = 256 threads = 8 waves. Tile: M=128 (wave w -> rows w*16..w*16+15),
// N=128 (8 x 16-wide accumulators per wave). K staged in chunks of 32 into a
// DOUBLE-BUFFERED LDS tile (row stride 136 dwords so the two K-halves of a
// fragment hit disjoint bank groups), filled by async global->LDS; tile t+1
// is prefetched while tile t is consumed (one s_wait_asynccnt + one barrier
// per chunk).
// A-frag (16x4 f32): lanes 0-15 row M=lane, K={k,k+1}; lanes 16-31 K={k+2,k+3}.
constexpr int KC   = 32;        // K chunk staged per barrier
constexpr int LDSW = 136;       // padded LDS row stride (dwords)

__global__ __launch_bounds__(256) void moe_gemm_k(
    const float* __restrict__ x, const float* __restrict__ We,
    const int* __restrict__ lists, const float* __restrict__ gws,
    const int* __restrict__ counts, float* __restrict__ out, int kp)
{
    const int e   = blockIdx.z;
    const int cnt = counts[kp * NE + e];
    const int m0  = blockIdx.y * 128;
    if (m0 >= cnt) return;                      // wave-uniform early exit

    const int tid  = threadIdx.x;
    const int wave = tid >> 5;                  // 0..7 -> M offset wave*16
    const int lane = tid & 31;
    const int hi   = lane >> 4;                 // K half (0/1)
    const int lo   = lane & 15;
    const int nwg  = blockIdx.x * 128;

    const int* __restrict__ lst = lists + (size_t)(kp * NE + e) * Tt;

    // gathered A row pointer for this lane (clamped for padded rows)
    const int rowA = m0 + wave * 16 + lo;
    const int tokA = lst[rowA < cnt ? rowA : (cnt - 1)];
    const float* pA = x + (size_t)tokA * Hd + hi * 2;

    // B global base for this expert / N block ([k][n] row-major)
    const float* pBg = We + (size_t)e * Hd * Hd + nwg;

    // staging map: thread copies rows (tid>>5)+8i, 16B chunk (tid&31)
    const int srow = tid >> 5;                  // 0..7
    const int scol = (tid & 31) * 4;            // dword column 0..124

    __shared__ float ldsB[2][KC * LDSW];

    auto stage = [&](int kc, int buf) {
#pragma unroll
        for (int i = 0; i < 4; ++i) {
            const int r = srow + 8 * i;
            const float* g = pBg + (size_t)(kc + r) * Hd + scol;
            float* l = &ldsB[buf][r * LDSW + scol];
#ifdef USE_ASYNC_LDS
            __builtin_amdgcn_global_load_async_to_lds_b128(GASV(g), LASV(l), 0, 0);
#else
            *(float4*)l = *(const float4*)g;    // b128 load + ds_store_b128
#endif
        }
    };

    v8f c[8] = {};

    constexpr int NT = Hd / KC;
    stage(0, 0);                                // prefetch tile 0
    for (int t = 0; t < NT; ++t) {
#ifdef USE_ASYNC_LDS
        __builtin_amdgcn_s_wait_asynccnt(0);    // my transfers for tile t done
#endif
        __syncthreads();                        // everyone's tile t visible;
                                                // everyone done reading t-1
        if (t + 1 < NT) stage((t + 1) * KC, (t + 1) & 1);

        const int kc = t * KC;
        const float* bufp = ldsB[t & 1];
#pragma unroll
        for (int kk = 0; kk < KC; kk += 4) {
            const v2f av = *(const v2f*)(pA + kc + kk);
            const float* lb = &bufp[(kk + hi * 2) * LDSW + lo];
#pragma unroll
            for (int j = 0; j < 8; ++j) {
                v2f bv;                         // B[k][n], B[k+1][n]
                bv.x = lb[j * 16];
                bv.y = lb[j * 16 + LDSW];
                c[j] = __builtin_amdgcn_wmma_f32_16x16x4_f32(
                           false, av, false, bv, (short)0, c[j], false, false);
            }
        }
    }

    // Epilogue: D VGPR jj holds row (jj + hi*8), col = nwg + j*16 + lo.
    // Non-atomic RMW is safe: each token occurs once per pass; passes are
    // serialized on the stream.
    const int rbase = m0 + wave * 16 + hi * 8;
#pragma unroll
    for (int jj = 0; jj < 8; ++jj) {
        const int r = rbase + jj;
        if (r < cnt) {
            const int tok = lst[r];
            const float g = gws[(size_t)kp * Tt + tok];
            float* po = out + (size_t)tok * Hd + nwg + lo;
#pragma unroll
            for (int j = 0; j < 8; ++j)
                po[j * 16] += g * c[j][jj];
        }
    }
}

// ------------------------------------------------------------------ launch --
extern "C" void kernel_launch(void* const* d_in, const int* in_sizes, int n_in,
                              void* d_out, int out_size, void* d_ws, size_t ws_size,
                              hipStream_t stream) {
    const float* x  = (const float*)d_in[0];   // (B,S,H) fp32
    const float* Wg = (const float*)d_in[1];   // (H,E)   fp32
    const float* We = (const float*)d_in[2];   // (E,H,H) fp32
    const float* be = (const float*)d_in[3];   // (E,H)   fp32
    float* out = (float*)d_out;                // (B,S,H) fp32

    // workspace layout
    char* ws = (char*)d_ws;
    int*   counts = (int*)ws;                                   // 16 ints
    int*   lists  = (int*)(ws + 256);                           // 2*E*T ints
    int*   topidx = (int*)(ws + 256 + (size_t)2 * NE * Tt * 4); // 2*T ints
    float* gws    = (float*)(ws + 256 + (size_t)2 * NE * Tt * 4
                                  + (size_t)2 * Tt * 4);        // 2*T floats

    (void)hipMemsetAsync(counts, 0, 2 * NE * sizeof(int), stream);

    moe_route_k<<<Tt / 8, 256, 0, stream>>>(x, Wg, counts, lists, topidx, gws);

    moe_bias_k<<<(Tt * (Hd / 4)) / 256, 256, 0, stream>>>(be, topidx, gws, out);

    dim3 grid(Hd / 128, Tt / 128, NE);
    moe_gemm_k<<<grid, 256, 0, stream>>>(x, We, lists, gws, counts, out, 0);
    moe_gemm_k<<<grid, 256, 0, stream>>>(x, We, lists, gws, counts, out, 1);
}